// SAGEencoder_84731114816418
// MI455X (gfx1250) — compile-verified
//
#include <hip/hip_runtime.h>

#define DIM 128  // in_channels == hidden_channels == 128

typedef __attribute__((ext_vector_type(2))) float v2f;
typedef __attribute__((ext_vector_type(8))) float v8f;

// ---------------------------------------------------------------------------
// Degree: cnt[i] = number of incoming edges (as float, for later 1/max(cnt,1))
// ---------------------------------------------------------------------------
__global__ void sage_degree_kernel(const int* __restrict__ dst,
                                   float* __restrict__ cnt, int E) {
    int e = blockIdx.x * blockDim.x + threadIdx.x;
    if (e < E) atomicAdd(&cnt[dst[e]], 1.0f);
}

// ---------------------------------------------------------------------------
// Scatter-add: agg[dst[e]] += x[src[e]].  One wave per edge; each of the 32
// lanes moves a float4 chunk of the 128-float row (coalesced 512B read),
// then 4 global_atomic_add_f32 into the destination row.
// ---------------------------------------------------------------------------
__global__ void sage_scatter_kernel(const float* __restrict__ x,
                                    const int* __restrict__ src,
                                    const int* __restrict__ dst,
                                    float* __restrict__ agg, int E) {
    long long tid = (long long)blockIdx.x * blockDim.x + threadIdx.x;
    int e = (int)(tid >> 5);
    if (e >= E) return;
    int lane = (int)(tid & 31);
    int s = src[e];
    int d = dst[e];
    const float4 v = *(const float4*)(x + (size_t)s * DIM + lane * 4);
    float* p = agg + (size_t)d * DIM + lane * 4;
    atomicAdd(p + 0, v.x);
    atomicAdd(p + 1, v.y);
    atomicAdd(p + 2, v.z);
    atomicAdd(p + 3, v.w);
}

// ---------------------------------------------------------------------------
// Fused SAGE linear stage (fp32 WMMA):
//   out[i,h] = (agg[i,:] * 1/max(cnt[i],1)) . Wl[h,:] + x[i,:] . Wr[h,:] + b[h]
// One wave -> one 16x16 output tile. Block = 8 waves = 16 rows x all 128 cols.
// K loop: 32 steps of K=4, 2 wmma per step (Wl term + Wr term) = 64 wmma/tile.
// ISA layouts (CDNA5 ISA):
//   A 16x4 f32 : lane l holds A[l%16, 2*(l/16)+{0,1}]  -> contiguous float2
//   B 4x16 f32 : lane l holds B[2*(l/16)+{0,1}, l%16] = W[l%16+col0, k..k+1]
//   C/D 16x16  : vgpr g, lane l -> (M = g + 8*(l/16), N = l%16)
// N (=50000) is an exact multiple of 16, so EXEC is all-ones (WMMA req).
// ---------------------------------------------------------------------------
__global__ __launch_bounds__(256)
void sage_gemm_kernel(const float* __restrict__ Agg,   // [N,128] un-normalized sums
                      const float* __restrict__ X,     // [N,128] self features
                      const float* __restrict__ Wl,    // [128,128] row-major
                      const float* __restrict__ Wr,    // [128,128] row-major
                      const float* __restrict__ bias,  // [128]
                      const float* __restrict__ cnt,   // [N] in-degree (float)
                      float* __restrict__ out,         // [N,128]
                      int relu) {
    const int wave = threadIdx.x >> 5;
    const int lane = threadIdx.x & 31;
    const int half = lane >> 4;   // 0: lanes 0-15, 1: lanes 16-31
    const int r    = lane & 15;
    const int row0 = blockIdx.x * 16;
    const int col0 = wave * 16;

    // mean normalization folded into the A operand
    const float inv = 1.0f / fmaxf(cnt[row0 + r], 1.0f);

    // C initialized with the bias: every output element (m,n) gets b[n]
    const float bv = bias[col0 + r];
    v8f c;
#pragma unroll
    for (int g = 0; g < 8; ++g) c[g] = bv;

    const float* aRow  = Agg + (size_t)(row0 + r) * DIM;
    const float* xRow  = X   + (size_t)(row0 + r) * DIM;
    const float* wlRow = Wl  + (size_t)(col0 + r) * DIM;
    const float* wrRow = Wr  + (size_t)(col0 + r) * DIM;

#pragma unroll 8
    for (int k = 0; k < DIM; k += 4) {
        const int kk = k + 2 * half;
        v2f a  = *(const v2f*)(aRow + kk);
        a = a * inv;  // scale aggregated sums -> mean
        v2f bl = *(const v2f*)(wlRow + kk);
        c = __builtin_amdgcn_wmma_f32_16x16x4_f32(false, a, false, bl,
                                                  (short)0, c, false, false);
        v2f xv = *(const v2f*)(xRow + kk);
        v2f br = *(const v2f*)(wrRow + kk);
        c = __builtin_amdgcn_wmma_f32_16x16x4_f32(false, xv, false, br,
                                                  (short)0, c, false, false);
    }

#pragma unroll
    for (int g = 0; g < 8; ++g) {
        float v = c[g];
        v = relu ? fmaxf(v, 0.0f) : v;
        out[(size_t)(row0 + g + 8 * half) * DIM + (col0 + r)] = v;
    }
}

// ---------------------------------------------------------------------------
// Launch: 3-layer GraphSAGE.  Layers 2 and 3 share the same aggregation of
// x1 (same edges, same input) -> only two scatter passes total.
// ---------------------------------------------------------------------------
extern "C" void kernel_launch(void* const* d_in, const int* in_sizes, int n_in,
                              void* d_out, int out_size, void* d_ws, size_t ws_size,
                              hipStream_t stream) {
    const float* x   = (const float*)d_in[0];
    const int*   ei  = (const int*)  d_in[1];   // [2, E] : row0 = src, row1 = dst
    const float* W1l = (const float*)d_in[2];
    const float* b1l = (const float*)d_in[3];
    const float* W1r = (const float*)d_in[4];
    const float* W2l = (const float*)d_in[5];
    const float* b2l = (const float*)d_in[6];
    const float* W2r = (const float*)d_in[7];
    const float* W3l = (const float*)d_in[8];
    const float* b3l = (const float*)d_in[9];
    const float* W3r = (const float*)d_in[10];

    const int N = in_sizes[0] / DIM;   // 50000
    const int E = in_sizes[1] / 2;     // 800000
    const int* src = ei;
    const int* dst = ei + E;

    const size_t NH = (size_t)N * DIM;

    // workspace layout: agg1 | x1 | agg2 | cnt  (~77 MB)
    float* agg1 = (float*)d_ws;
    float* x1   = agg1 + NH;
    float* agg2 = x1 + NH;
    float* cnt  = agg2 + NH;

    float* h_out1 = (float*)d_out;        // h_ from layer 2
    float* h_out2 = h_out1 + NH;          // h  from layer 3

    // zero accumulators every call (graph replays re-run the atomics)
    hipMemsetAsync(agg1, 0, NH * sizeof(float), stream);
    hipMemsetAsync(agg2, 0, NH * sizeof(float), stream);
    hipMemsetAsync(cnt,  0, (size_t)N * sizeof(float), stream);

    // degree (shared by all layers)
    sage_degree_kernel<<<(E + 255) / 256, 256, 0, stream>>>(dst, cnt, E);

    const long long scatterThreads = (long long)E * 32;
    const int scatterBlocks = (int)((scatterThreads + 255) / 256);
    const int gemmBlocks = N / 16;  // 3125 (exact)

    // ---- layer 1: x1 = relu(mean_agg(x) @ W1l^T + b1l + x @ W1r^T)
    sage_scatter_kernel<<<scatterBlocks, 256, 0, stream>>>(x, src, dst, agg1, E);
    sage_gemm_kernel<<<gemmBlocks, 256, 0, stream>>>(agg1, x, W1l, W1r, b1l, cnt, x1, 1);

    // ---- shared aggregation of x1 for layers 2 and 3
    sage_scatter_kernel<<<scatterBlocks, 256, 0, stream>>>(x1, src, dst, agg2, E);

    // ---- layer 2: h_ ; layer 3: h
    sage_gemm_kernel<<<gemmBlocks, 256, 0, stream>>>(agg2, x1, W2l, W2r, b2l, cnt, h_out1, 0);
    sage_gemm_kernel<<<gemmBlocks, 256, 0, stream>>>(agg2, x1, W3l, W3r, b3l, cnt, h_out2, 0);
}